// MessagePassing_15040975470795
// MI455X (gfx1250) — compile-verified
//
#include <hip/hip_runtime.h>

// Graph std-aggregation message passing (PyG aggregator='std'):
//   out[n,f] = std over {x[src[e],f] : tgt[e]==n}, zeroed where count<=1
//
// One edge pass using the moment identity sum((x-m)^2) = sum(x^2) - c*m^2:
//   pass 1 (edges): atomically accumulate count[t], sum[t,:], sumsq[t,:]
//   pass 2 (nodes): mean = sum/max(c,eps); var = sumsq/max(c,eps) - mean^2
//                   std = sqrt(max(var,0)); std = 0 if c <= 1
//
// MI455X reasoning: x (25.6MB) and accumulators (51.6MB) are L2-resident
// (192MB L2), so gathers and atomics run at L2 speed; HBM traffic ~100MB
// (~5us at 23.3TB/s). The binding resource is L2 atomic-f32-add throughput,
// so we minimize atomics (one pass, 2 per gathered float = minimum for a
// moment-based std) and use native wave32 f32 global atomics (relaxed agent
// scope -> global_atomic_add_f32, no CAS). Edge indices are streamed once ->
// non-temporal loads + global_prefetch_b8; output is write-once -> NT store.

#define NFEAT 64
#define FEAT4 (NFEAT / 4)     // 16 float4 chunks per 64-float feature row
#define EPS 1e-8f

// Native vector type (required by __builtin_nontemporal_load/store).
typedef float f32x4 __attribute__((ext_vector_type(4)));

__device__ __forceinline__ void atomic_add_f32_agent(float* p, float v) {
    // Lowers to global_atomic_add_f32 (no return, relaxed, device scope)
    __hip_atomic_fetch_add(p, v, __ATOMIC_RELAXED, __HIP_MEMORY_SCOPE_AGENT);
}

__global__ __launch_bounds__(256) void edge_scatter_kernel(
    const float* __restrict__ x,
    const int* __restrict__ src_idx,
    const int* __restrict__ tgt_idx,
    float* __restrict__ sum,
    float* __restrict__ sumsq,
    unsigned int* __restrict__ count,
    int n_edges)
{
    // 32-bit thread id: n_edges * 16 = 25.6M < 2^31
    const int tid = blockIdx.x * 256 + threadIdx.x;
    const int e = tid >> 4;          // 16 threads cooperate on one edge
    const int q = tid & 15;          // which float4 of the 64-float row
    if (e >= n_edges) return;

    // Prefetch the streaming edge-index arrays ahead (global_prefetch_b8).
    if (q == 0) {
        const int pf = e + 8192;
        if (pf < n_edges) {
            __builtin_prefetch(&src_idx[pf], 0, 1);
            __builtin_prefetch(&tgt_idx[pf], 0, 1);
        }
    }

    // Streamed exactly once -> non-temporal (TH=NT) loads keep L2 for the
    // hot x/sum/sumsq working set.
    const int s = __builtin_nontemporal_load(&src_idx[e]);
    const int t = __builtin_nontemporal_load(&tgt_idx[e]);

    // Coalesced 256B gather per 16-lane half-wave: global_load_b128 per lane.
    const f32x4 v = *(const f32x4*)(x + (s * NFEAT + q * 4));

    float* sp = sum   + (t * NFEAT + q * 4);
    float* qp = sumsq + (t * NFEAT + q * 4);

    atomic_add_f32_agent(sp + 0, v.x);
    atomic_add_f32_agent(sp + 1, v.y);
    atomic_add_f32_agent(sp + 2, v.z);
    atomic_add_f32_agent(sp + 3, v.w);

    atomic_add_f32_agent(qp + 0, v.x * v.x);
    atomic_add_f32_agent(qp + 1, v.y * v.y);
    atomic_add_f32_agent(qp + 2, v.z * v.z);
    atomic_add_f32_agent(qp + 3, v.w * v.w);

    if (q == 0) {
        atomicAdd(&count[t], 1u);   // native global_atomic_add_u32
    }
}

__global__ __launch_bounds__(256) void node_finalize_kernel(
    const float* __restrict__ sum,
    const float* __restrict__ sumsq,
    const unsigned int* __restrict__ count,
    float* __restrict__ out,
    int n_nodes)
{
    const int tid = blockIdx.x * 256 + threadIdx.x;
    const int n = tid >> 4;
    const int q = tid & 15;
    if (n >= n_nodes) return;

    const unsigned int c = count[n];
    const float cf = (float)c;
    const float inv = 1.0f / fmaxf(cf, EPS);
    const bool zero_out = (c <= 1u);

    const f32x4 s  = *(const f32x4*)(sum   + (n * NFEAT + q * 4));
    const f32x4 ss = *(const f32x4*)(sumsq + (n * NFEAT + q * 4));

    f32x4 r;
    {
        float m = s.x * inv; float var = fmaxf(ss.x * inv - m * m, 0.0f);
        r.x = zero_out ? 0.0f : sqrtf(var);
    }
    {
        float m = s.y * inv; float var = fmaxf(ss.y * inv - m * m, 0.0f);
        r.y = zero_out ? 0.0f : sqrtf(var);
    }
    {
        float m = s.z * inv; float var = fmaxf(ss.z * inv - m * m, 0.0f);
        r.z = zero_out ? 0.0f : sqrtf(var);
    }
    {
        float m = s.w * inv; float var = fmaxf(ss.w * inv - m * m, 0.0f);
        r.w = zero_out ? 0.0f : sqrtf(var);
    }

    // Output is write-once -> non-temporal b128 store (TH=NT).
    __builtin_nontemporal_store(r, (f32x4*)(out + (n * NFEAT + q * 4)));
}

extern "C" void kernel_launch(void* const* d_in, const int* in_sizes, int n_in,
                              void* d_out, int out_size, void* d_ws, size_t ws_size,
                              hipStream_t stream) {
    const float* x          = (const float*)d_in[0];
    const int*   edge_index = (const int*)d_in[1];

    const int n_nodes = in_sizes[0] / NFEAT;    // 100000
    const int n_edges = in_sizes[1] / 2;        // 1600000

    const int* src_idx = edge_index;            // edge_index[0, :]
    const int* tgt_idx = edge_index + n_edges;  // edge_index[1, :]

    // Workspace layout (256B aligned regions):
    //   [0)          count : n_nodes u32
    //   [sum_off)    sum   : n_nodes*64 f32
    //   [sumsq_off)  sumsq : n_nodes*64 f32
    const size_t count_bytes = (size_t)n_nodes * sizeof(unsigned int);
    const size_t sum_off     = (count_bytes + 255) & ~(size_t)255;
    const size_t feat_bytes  = (size_t)n_nodes * NFEAT * sizeof(float);
    const size_t sumsq_off   = (sum_off + feat_bytes + 255) & ~(size_t)255;
    const size_t total_bytes = sumsq_off + feat_bytes;

    unsigned int* count = (unsigned int*)d_ws;
    float*        sum   = (float*)((char*)d_ws + sum_off);
    float*        sumsq = (float*)((char*)d_ws + sumsq_off);
    float*        out   = (float*)d_out;

    // Zero accumulators every call (graph-capture legal, deterministic).
    (void)hipMemsetAsync(d_ws, 0, total_bytes, stream);

    {
        const int threads = n_edges * FEAT4;           // 25.6M
        const int block = 256;
        const int grid = (threads + block - 1) / block;
        edge_scatter_kernel<<<grid, block, 0, stream>>>(
            x, src_idx, tgt_idx, sum, sumsq, count, n_edges);
    }
    {
        const int threads = n_nodes * FEAT4;           // 1.6M
        const int block = 256;
        const int grid = (threads + block - 1) / block;
        node_finalize_kernel<<<grid, block, 0, stream>>>(
            sum, sumsq, count, out, n_nodes);
    }
}